// LightGCN_3539053052410
// MI455X (gfx1250) — compile-verified
//
#include <hip/hip_runtime.h>

#ifndef __has_builtin
#define __has_builtin(x) 0
#endif

#if __has_builtin(__builtin_amdgcn_global_load_async_to_lds_b128) && \
    __has_builtin(__builtin_amdgcn_s_wait_asynccnt)
#define USE_ASYNC_GATHER 1
#else
#define USE_ASYNC_GATHER 0
#endif

#define TPB 256
#define ROW4 16   // 64 floats per row = 16 float4
#define EPT 4     // edges per 16-lane group (pipeline depth via double buffer)

// Matches the builtin's parameter type (probe-confirmed in round 1):
// '__attribute__((__vector_size__(4 * sizeof(int)))) int __device__ *'
typedef int v4i __attribute__((vector_size(16)));

#if USE_ASYNC_GATHER
#define ASYNC_LD_B128(gsrc, ldst)                                      \
    __builtin_amdgcn_global_load_async_to_lds_b128(                    \
        (__attribute__((address_space(1))) v4i*)(v4i*)(gsrc),          \
        (__attribute__((address_space(3))) v4i*)(v4i*)(ldst),          \
        /*offset=*/0, /*cpol=*/0)
#endif

__device__ __forceinline__ int imin(int a, int b) { return a < b ? a : b; }

// ---------------------------------------------------------------- utilities
__global__ void k_zero(float* __restrict__ p, long n) {
    long i = (long)blockIdx.x * blockDim.x + threadIdx.x;
    if (i < n) p[i] = 0.f;
}

// deg[dst] += 1  (float segment_sum of ones, matches reference)
__global__ void k_deg(const int* __restrict__ dst, float* __restrict__ deg, int E) {
    int e = blockIdx.x * blockDim.x + threadIdx.x;
    if (e < E) atomicAdd(&deg[dst[e]], 1.0f);
}

// in-place: deg -> deg^{-1/2} (0 where deg == 0)
__global__ void k_dinv(float* __restrict__ deg, int N) {
    int i = blockIdx.x * blockDim.x + threadIdx.x;
    if (i < N) {
        float d = deg[i];
        deg[i] = (d > 0.f) ? rsqrtf(d) : 0.f;
    }
}

// norm[e] = dinv[src] * dinv[dst]
__global__ void k_norm(const int* __restrict__ src, const int* __restrict__ dst,
                       const float* __restrict__ dinv, float* __restrict__ norm, int E) {
    int e = blockIdx.x * blockDim.x + threadIdx.x;
    if (e < E) norm[e] = dinv[src[e]] * dinv[dst[e]];
}

// cur = out = concat(users, items); zero layer-0 scatter target (bufB)
__global__ void k_init(const float4* __restrict__ users, const float4* __restrict__ items,
                       float4* __restrict__ cur, float4* __restrict__ zbuf,
                       float4* __restrict__ out, long nu4, long n4) {
    long i = (long)blockIdx.x * blockDim.x + threadIdx.x;
    if (i < n4) {
        float4 v = (i < nu4) ? users[i] : items[i - nu4];
        cur[i]  = v;
        out[i]  = v;
        zbuf[i] = make_float4(0.f, 0.f, 0.f, 0.f);
    }
}

// One propagation layer: nxt[dst] += norm[e] * cur[src].
// Block = 16 groups of 16 lanes; a group handles EPT edges, one 256B row per
// edge (lane -> one float4). Async global->LDS staging is double-buffered so
// the gather for edge j+1 is in flight while edge j's atomics issue. Edge
// indices are clamped (not branched) so the wave-level ASYNCcnt issue/wait
// pattern is uniform; only the consume side is masked by e < E.
__global__ void k_scatter(const int* __restrict__ src, const int* __restrict__ dst,
                          const float* __restrict__ norm,
                          const float4* __restrict__ cur, float* __restrict__ nxt,
                          int E) {
    const int c     = threadIdx.x & 15;        // float4 slot within the row
    const int group = threadIdx.x >> 4;        // 0..15
    const int base  = blockIdx.x * (16 * EPT); // first edge of this block

#if USE_ASYNC_GATHER
    __shared__ float4 stage[2][TPB];

    {   // prologue: issue gather for j = 0
        int e0 = imin(base + group, E - 1);
        ASYNC_LD_B128(cur + ((size_t)src[e0] * ROW4 + c), &stage[0][threadIdx.x]);
    }

    for (int j = 0; j < EPT; ++j) {
        int e = base + j * 16 + group;
        if (j + 1 < EPT) {                        // uniform condition
            int en = imin(e + 16, E - 1);
            ASYNC_LD_B128(cur + ((size_t)src[en] * ROW4 + c),
                          &stage[(j + 1) & 1][threadIdx.x]);
            __builtin_amdgcn_s_wait_asynccnt(1);  // buffer j has landed
        } else {
            __builtin_amdgcn_s_wait_asynccnt(0);
        }
        if (e < E) {
            float  w = norm[e];
            int    d = dst[e];
            float4 v = stage[j & 1][threadIdx.x];
            float* o = nxt + ((size_t)d * 64 + (size_t)c * 4);
            atomicAdd(o + 0, w * v.x);
            atomicAdd(o + 1, w * v.y);
            atomicAdd(o + 2, w * v.z);
            atomicAdd(o + 3, w * v.w);
        }
    }
#else
    for (int j = 0; j < EPT; ++j) {
        int e = base + j * 16 + group;
        if (e < E) {
            float  w = norm[e];
            int    d = dst[e];
            float4 v = cur[(size_t)src[e] * ROW4 + c];
            float* o = nxt + ((size_t)d * 64 + (size_t)c * 4);
            atomicAdd(o + 0, w * v.x);
            atomicAdd(o + 1, w * v.y);
            atomicAdd(o + 2, w * v.z);
            atomicAdd(o + 3, w * v.w);
        }
    }
#endif
}

// out = (out + nxt) * scale, and zero `dead` (the just-consumed cur buffer,
// which becomes the next layer's scatter target) in the same streaming pass.
__global__ void k_acc(const float4* __restrict__ nxt, float4* __restrict__ out,
                      float4* __restrict__ dead, float scale, long n4) {
    long i = (long)blockIdx.x * blockDim.x + threadIdx.x;
    if (i < n4) {
        float4 a = out[i];
        float4 b = nxt[i];
        a.x = (a.x + b.x) * scale;
        a.y = (a.y + b.y) * scale;
        a.z = (a.z + b.z) * scale;
        a.w = (a.w + b.w) * scale;
        out[i]  = a;
        dead[i] = make_float4(0.f, 0.f, 0.f, 0.f);
    }
}

// ---------------------------------------------------------------- launcher
extern "C" void kernel_launch(void* const* d_in, const int* in_sizes, int n_in,
                              void* d_out, int out_size, void* d_ws, size_t ws_size,
                              hipStream_t stream) {
    const float* users = (const float*)d_in[0];
    const float* items = (const float*)d_in[1];
    const int*   edge  = (const int*)d_in[2];

    const long nuD = in_sizes[0];           // num_users * 64
    const long niD = in_sizes[1];           // num_items * 64
    const int  E   = in_sizes[2] / 2;
    const int  num_users = (int)(nuD / 64);
    const int  num_items = (int)(niD / 64);
    const int  N   = num_users + num_items;
    const long nD  = (long)N * 64;
    const long n4  = nD / 4;                // float4 count
    const long nu4 = nuD / 4;

    const int* src = edge;                  // edge_index row 0
    const int* dst = edge + E;              // edge_index row 1

    // workspace partition (N*4 and E*4 bytes are multiples of 16 -> aligned)
    float* deg  = (float*)d_ws;             // N floats (becomes dinv in-place)
    float* nrm  = deg + N;                  // E floats
    float* bufA = nrm + E;                  // N*64 floats
    float* bufB = bufA + nD;                // N*64 floats

    float* out = (float*)d_out;

    const int  gN  = (N + TPB - 1) / TPB;
    const int  gE  = (E + TPB - 1) / TPB;
    const long gn4 = (n4 + TPB - 1) / TPB;
    const int  gS  = (E + 16 * EPT - 1) / (16 * EPT);  // blocks of 64 edges

    // degree -> dinv -> edge norms
    k_zero <<<gN, TPB, 0, stream>>>(deg, N);
    k_deg  <<<gE, TPB, 0, stream>>>(dst, deg, E);
    k_dinv <<<gN, TPB, 0, stream>>>(deg, N);
    k_norm <<<gE, TPB, 0, stream>>>(src, dst, deg, nrm, E);

    // acc (= d_out), layer-0 embeddings in bufA, zeroed scatter target bufB
    k_init <<<gn4, TPB, 0, stream>>>((const float4*)users, (const float4*)items,
                                     (float4*)bufA, (float4*)bufB, (float4*)out,
                                     nu4, n4);

    float* cur = bufA;
    float* nxt = bufB;
    const float inv25 = 1.0f / 25.0f;       // 1/(L+1)^2

    for (int layer = 0; layer < 4; ++layer) {
        k_scatter <<<gS, TPB, 0, stream>>>(src, dst, nrm, (const float4*)cur, nxt, E);
        // accumulate nxt into out; zero cur (it becomes the next scatter target)
        k_acc     <<<gn4, TPB, 0, stream>>>((const float4*)nxt, (float4*)out,
                                            (float4*)cur,
                                            (layer == 3) ? inv25 : 1.0f, n4);
        float* tmp = cur; cur = nxt; nxt = tmp;
    }
}